// Block_34162169872771
// MI455X (gfx1250) — compile-verified
//
#include <hip/hip_runtime.h>
#include <math.h>

typedef __attribute__((ext_vector_type(2))) float v2f;
typedef __attribute__((ext_vector_type(4))) float v4f;
typedef __attribute__((ext_vector_type(8))) float v8f;

#define C_DIM 2048
#define T_DIM 4096
#define K_DIM 2048

// ---------------------------------------------------------------------------
// Tiled fp32 WMMA GEMM:  Out[i,t] = sum_j W[i,j] * f(In[j,t])   (+ epilogue)
//   M = 2048, N = 4096, K = 2048.
//   Block tile 128x64, BK=32, 256 threads = 8 waves; each wave owns a 16x64
//   strip (4 f32 accumulators), so one A fragment feeds 4 WMMAs.
//   B is stored K-pair-interleaved in LDS: element (k,n) at Bs[k>>1][2n+(k&1)]
//   -> every WMMA B fragment is a single aligned ds_load_b64, no packing movs.
//   Pair-row stride 160 words == 32 (mod 64 banks): the two K-half lane groups
//   hit disjoint bank halves -> conflict-free.
// ---------------------------------------------------------------------------
template <bool RELU_IN, bool TANH_EPI>
__global__ __launch_bounds__(256) void gemm128(const float* __restrict__ W,
                                               const float* __restrict__ In,
                                               float* __restrict__ Out,
                                               const float* __restrict__ Orig,
                                               const float* __restrict__ gains) {
  constexpr int BM = 128, BN = 64, BK = 32;
  __shared__ float As[BM][BK + 4];   // [m][k], row stride 36 words (144 B)
  __shared__ float Bs[BK / 2][2 * BN + 32];  // pair-interleaved, stride 160 words

  const int tid  = threadIdx.x;
  const int i0   = blockIdx.y * BM;  // output row tile
  const int t0   = blockIdx.x * BN;  // output col (frame) tile
  const int lane = tid & 31;
  const int wid  = tid >> 5;
  const int mb   = wid * 16;         // wave's M strip inside block tile
  const int col  = lane & 15;        // N (and A-row) index within fragment
  const int kh   = lane >> 4;        // K-half select per ISA layout

  // A tile: 128 rows x 32 cols, 4 float4 per thread (half a row each)
  const int ar  = tid >> 1;          // 0..127
  const int aq0 = (tid & 1) * 16;    // 0 or 16
  // B tile: 16 pair-rows x 64 cols, 2 float4 global loads per thread
  const int pr = tid >> 4;           // pair row 0..15 (k = 2*pr, 2*pr+1)
  const int bn = (tid & 15) * 4;     // n start

  v8f acc0 = {}, acc1 = {}, acc2 = {}, acc3 = {};

  for (int k0 = 0; k0 < K_DIM; k0 += BK) {
    // ---- stage A tile ----
    const float* wp = W + (size_t)(i0 + ar) * K_DIM + k0 + aq0;
#pragma unroll
    for (int j = 0; j < 4; ++j)
      *(v4f*)&As[ar][aq0 + 4 * j] = *(const v4f*)(wp + 4 * j);

    // ---- stage B tile (K-pair interleaved) ----
    const float* ip = In + (size_t)(k0 + 2 * pr) * T_DIM + t0 + bn;
    v4f e = *(const v4f*)ip;            // k = 2*pr
    v4f o = *(const v4f*)(ip + T_DIM);  // k = 2*pr + 1
    if constexpr (RELU_IN) {
#pragma unroll
      for (int j = 0; j < 4; ++j) {
        e[j] = fmaxf(e[j], 0.0f);
        o[j] = fmaxf(o[j], 0.0f);
      }
    }
    v4f lo = {e[0], o[0], e[1], o[1]};
    v4f hi = {e[2], o[2], e[3], o[3]};
    *(v4f*)&Bs[pr][2 * bn]     = lo;
    *(v4f*)&Bs[pr][2 * bn + 4] = hi;

    __syncthreads();

#pragma unroll
    for (int kk = 0; kk < BK; kk += 4) {
      const int p = (kk >> 1) + kh;  // B pair-row for this lane's K-half
      // A fragment: A[M=col][K = kk+2*kh, +1]  (aligned ds_load_b64)
      v2f a = *(const v2f*)&As[mb + col][kk + 2 * kh];
      // B fragments for the 4 N-subtiles: single b64 each, mergeable in pairs
      v2f b0 = *(const v2f*)&Bs[p][2 * col];
      v2f b1 = *(const v2f*)&Bs[p][2 * (col + 16)];
      v2f b2 = *(const v2f*)&Bs[p][2 * (col + 32)];
      v2f b3 = *(const v2f*)&Bs[p][2 * (col + 48)];
      acc0 = __builtin_amdgcn_wmma_f32_16x16x4_f32(false, a, false, b0,
                                                   (short)0, acc0, false, false);
      acc1 = __builtin_amdgcn_wmma_f32_16x16x4_f32(false, a, false, b1,
                                                   (short)0, acc1, false, false);
      acc2 = __builtin_amdgcn_wmma_f32_16x16x4_f32(false, a, false, b2,
                                                   (short)0, acc2, false, false);
      acc3 = __builtin_amdgcn_wmma_f32_16x16x4_f32(false, a, false, b3,
                                                   (short)0, acc3, false, false);
    }
    __syncthreads();
  }

  // C/D layout: VGPR v holds (M = v + 8*kh, N = col)
  const int    gr0 = i0 + mb + kh * 8;
  const size_t gc  = (size_t)t0 + col;
#pragma unroll
  for (int v = 0; v < 8; ++v) {
    const int    r    = gr0 + v;
    const size_t base = (size_t)r * T_DIM + gc;
    if constexpr (TANH_EPI) {
      const float gg = 5.0f / (1.0f + expf(-gains[r]));  // sigmoid(g)*MAX_GAIN
      Out[base]      = tanhf(gg * (acc0[v] + Orig[base]));
      Out[base + 16] = tanhf(gg * (acc1[v] + Orig[base + 16]));
      Out[base + 32] = tanhf(gg * (acc2[v] + Orig[base + 32]));
      Out[base + 48] = tanhf(gg * (acc3[v] + Orig[base + 48]));
    } else {
      Out[base]      = acc0[v];
      Out[base + 16] = acc1[v];
      Out[base + 32] = acc2[v];
      Out[base + 48] = acc3[v];
    }
  }
}

// ---------------------------------------------------------------------------
// The fft_convolve with denv[c,t] = d^(t+1) is exactly the causal IIR
//   y[t] = d * (x[t] + y[t-1]),  y[-1] = 0.
// One wave per channel; 32-frame segments; intra-wave Hillis-Steele scan of
// affine transforms (m, a): y = m*y_prev + a, element transform (d, d*x).
// ---------------------------------------------------------------------------
__global__ __launch_bounds__(256) void iir_scan(const float* __restrict__ X,
                                                const float* __restrict__ decays,
                                                float* __restrict__ Y) {
  const int lane = threadIdx.x & 31;
  const int ch   = blockIdx.x * 8 + (threadIdx.x >> 5);  // 256 blocks * 8 waves
  const float d  = 0.5f + 0.5f / (1.0f + expf(-decays[ch]));  // base + sigmoid*span
  const float* x = X + (size_t)ch * T_DIM;
  float*       y = Y + (size_t)ch * T_DIM;

  float carry = 0.0f;
  for (int s = 0; s < T_DIM; s += 32) {
    float A = d * x[s + lane];
    float M = d;
#pragma unroll
    for (int off = 1; off < 32; off <<= 1) {
      float Ap = __shfl_up(A, (unsigned)off, 32);
      float Mp = __shfl_up(M, (unsigned)off, 32);
      if (lane >= off) {
        A = fmaf(M, Ap, A);
        M *= Mp;
      }
    }
    const float yv = fmaf(M, carry, A);  // lane l: d^(l+1)*carry + partial
    y[s + lane] = yv;
    carry = __shfl(yv, 31, 32);
  }
}

// ---------------------------------------------------------------------------
// tmp[w][t] (2048x4096) -> out[t][w] (4096x2048): time-major audio output.
// ---------------------------------------------------------------------------
__global__ __launch_bounds__(256) void transpose_wt(const float* __restrict__ in,
                                                    float* __restrict__ out) {
  __shared__ float tile[32][33];
  const int tx = threadIdx.x;            // 0..31
  const int ty = threadIdx.y;            // 0..7
  const int t0 = blockIdx.x * 32;
  const int w0 = blockIdx.y * 32;
#pragma unroll
  for (int r = ty; r < 32; r += 8)
    tile[r][tx] = in[(size_t)(w0 + r) * T_DIM + t0 + tx];
  __syncthreads();
#pragma unroll
  for (int r = ty; r < 32; r += 8)
    out[(size_t)(t0 + r) * C_DIM + w0 + tx] = tile[tx][r];
}

// ---------------------------------------------------------------------------
extern "C" void kernel_launch(void* const* d_in, const int* in_sizes, int n_in,
                              void* d_out, int out_size, void* d_ws, size_t ws_size,
                              hipStream_t stream) {
  const float* cp     = (const float*)d_in[0];  // (1, 2048, 4096)
  const float* w1     = (const float*)d_in[1];  // (2048, 2048)
  const float* w2     = (const float*)d_in[2];  // (2048, 2048)
  const float* audio  = (const float*)d_in[3];  // (2048, 2048)
  const float* decays = (const float*)d_in[4];  // (2048,)
  const float* gains  = (const float*)d_in[5];  // (2048,)

  float* out       = (float*)d_out;
  float* audio_out = out;                                   // 4096*2048 floats
  float* cp_out    = out + (size_t)T_DIM * C_DIM;           // 2048*4096 floats

  float* X1  = (float*)d_ws;                                // 32 MB
  float* Y   = X1 + (size_t)C_DIM * T_DIM;                  // 32 MB
  float* tmp = X1;  // X1 dead after GEMM2 -> reuse for pre-transpose audio

  const dim3 gemmGrid(T_DIM / 64, C_DIM / 128);  // (64, 16)
  const dim3 gemmBlock(256);

  // 1) X1 = w1 @ relu(cp)
  gemm128<true, false><<<gemmGrid, gemmBlock, 0, stream>>>(w1, cp, X1, nullptr, nullptr);

  // 2) Y[c,t] = d[c] * (X1[c,t] + Y[c,t-1])   (== fft_convolve with d^(t+1))
  iir_scan<<<C_DIM / 8, 256, 0, stream>>>(X1, decays, Y);

  // 3) cp_out = tanh(sigmoid(gains)*5 * (w2 @ Y + X1))
  gemm128<false, true><<<gemmGrid, gemmBlock, 0, stream>>>(w2, Y, cp_out, X1, gains);

  // 4) tmp[w,t] = audio @ cp_out
  gemm128<false, false><<<gemmGrid, gemmBlock, 0, stream>>>(audio, cp_out, tmp, nullptr, nullptr);

  // 5) audio_out[t,w] = tmp[w,t]
  transpose_wt<<<dim3(T_DIM / 32, C_DIM / 32), dim3(32, 8), 0, stream>>>(tmp, audio_out);
}